// NHGCFLayer_65910568124540
// MI455X (gfx1250) — compile-verified
//
#include <hip/hip_runtime.h>

typedef __attribute__((ext_vector_type(2))) float v2f;
typedef __attribute__((ext_vector_type(8))) float v8f;

#define D64 64

// ---------------------------------------------------------------------------
// Fused GCN dense part: for rows r of x = concat(x1, x2):
//   t[r] = x[r] @ Tw + Tb          -> out_t  (also serves as init for +t term)
//   p[r] = t[r] + (x[r]*x[r]) @ Iw + Ib -> out_p (edge gather source)
// One block = 16 rows x 64 cols, 4 waves, each wave owns a 16-col tile.
// Uses V_WMMA_F32_16X16X4_F32, 16 K-steps, two accumulator chains.
// ---------------------------------------------------------------------------
__global__ __launch_bounds__(128)
void gcn_gemm_kernel(const float* __restrict__ x1, int n1,
                     const float* __restrict__ x2, int n2,
                     const float* __restrict__ Tw, const float* __restrict__ Tb,
                     const float* __restrict__ Iw, const float* __restrict__ Ib,
                     float* __restrict__ out_t, float* __restrict__ out_p)
{
    const int wave  = threadIdx.x >> 5;          // 0..3 -> column tile
    const int lane  = threadIdx.x & 31;
    const int m     = lane & 15;                 // row (A) / col (B,C) within tile
    const int khalf = (lane >> 4) * 2;           // lanes 16-31 hold K=2,3
    const int rowBase = blockIdx.x * 16;
    const int colBase = wave * 16;

    const int row = rowBase + m;
    const float* xr = (row < n1) ? (x1 + (size_t)row * D64)
                                 : (x2 + (size_t)(row - n1) * D64);
    const int col = colBase + m;

    v8f ct = {};   // t accumulator (16x16 f32, 8 VGPRs)
    v8f cq = {};   // q accumulator

    #pragma unroll
    for (int k0 = 0; k0 < D64; k0 += 4) {
        const int kk = k0 + khalf;
        const float a0 = xr[kk];
        const float a1 = xr[kk + 1];
        v2f aT; aT.x = a0;      aT.y = a1;
        v2f aQ; aQ.x = a0 * a0; aQ.y = a1 * a1;
        v2f bT; bT.x = Tw[kk * D64 + col]; bT.y = Tw[(kk + 1) * D64 + col];
        v2f bQ; bQ.x = Iw[kk * D64 + col]; bQ.y = Iw[(kk + 1) * D64 + col];
        ct = __builtin_amdgcn_wmma_f32_16x16x4_f32(false, aT, false, bT,
                                                   (short)0, ct, false, false);
        cq = __builtin_amdgcn_wmma_f32_16x16x4_f32(false, aQ, false, bQ,
                                                   (short)0, cq, false, false);
    }

    // C/D layout: VGPR v -> M = v + (lane<16 ? 0 : 8), N = lane&15
    const int mrow0 = rowBase + ((lane >> 4) << 3);
    const float tb = Tb[col];
    const float ib = Ib[col];
    #pragma unroll
    for (int v = 0; v < 8; ++v) {
        const float tval = ct[v] + tb;
        const float qval = cq[v] + ib;
        const size_t o = (size_t)(mrow0 + v) * D64 + col;
        out_t[o] = tval;
        out_p[o] = tval + qval;   // t + q : single gather row for edge kernel
    }
}

// ---------------------------------------------------------------------------
// Edge scatter: out[row] += val * p[col]   (p = t + q, out pre-initialized = t)
// 4 threads per edge, float4 gathers, hardware f32 global atomics.
// ---------------------------------------------------------------------------
__global__ __launch_bounds__(256)
void edge_scatter_kernel(const int* __restrict__ rows, const int* __restrict__ cols,
                         const float* __restrict__ vals, int nnz,
                         const float* __restrict__ p, float* __restrict__ out)
{
    const int gid = blockIdx.x * blockDim.x + threadIdx.x;
    const int e = gid >> 2;
    if (e >= nnz) return;
    const int q = gid & 3;
    const int r = rows[e];
    const int c = cols[e];
    const float v = vals[e];
    const float4* ps = (const float4*)(p + (size_t)c * D64 + q * 16);
    float* ob = out + (size_t)r * D64 + q * 16;
    #pragma unroll
    for (int i = 0; i < 4; ++i) {
        const float4 pv = ps[i];
        unsafeAtomicAdd(ob + 4 * i + 0, v * pv.x);
        unsafeAtomicAdd(ob + 4 * i + 1, v * pv.y);
        unsafeAtomicAdd(ob + 4 * i + 2, v * pv.z);
        unsafeAtomicAdd(ob + 4 * i + 3, v * pv.w);
    }
}

// ---------------------------------------------------------------------------
// Relational attention over 3 relations: one wave32 per node, lane = hidden h
// (H == 32 == wave width). w_r = sum_h tanh(z_r . w1[:,h] + b1[h]) * w2[h];
// softmax over r; out = sum_r beta_r * z_r.
// ---------------------------------------------------------------------------
__global__ __launch_bounds__(256)
void rel_att_kernel(const float* __restrict__ z0, const float* __restrict__ z1,
                    const float* __restrict__ z2, int n,
                    const float* __restrict__ w1, const float* __restrict__ b1,
                    const float* __restrict__ w2, float* __restrict__ outp)
{
    const int H = 32;
    const int wave = threadIdx.x >> 5;
    const int lane = threadIdx.x & 31;
    const int node = blockIdx.x * 8 + wave;
    if (node >= n) return;

    const float* zr[3] = { z0 + (size_t)node * D64,
                           z1 + (size_t)node * D64,
                           z2 + (size_t)node * D64 };
    const float bh  = b1[lane];
    const float w2h = w2[lane];

    float w[3];
    #pragma unroll
    for (int r = 0; r < 3; ++r) {
        float acc = bh;
        #pragma unroll 8
        for (int d = 0; d < D64; ++d)
            acc += zr[r][d] * w1[d * H + lane];
        float t = tanhf(acc) * w2h;
        #pragma unroll
        for (int off = 16; off > 0; off >>= 1)    // wave32 butterfly reduce
            t += __shfl_xor(t, off, 32);
        w[r] = t;
    }

    const float mx = fmaxf(w[0], fmaxf(w[1], w[2]));
    const float e0 = __expf(w[0] - mx);
    const float e1 = __expf(w[1] - mx);
    const float e2 = __expf(w[2] - mx);
    const float inv = 1.0f / (e0 + e1 + e2);
    const float be0 = e0 * inv, be1 = e1 * inv, be2 = e2 * inv;

    #pragma unroll
    for (int d = lane; d < D64; d += 32)
        outp[(size_t)node * D64 + d] = be0 * zr[0][d] + be1 * zr[1][d] + be2 * zr[2][d];
}

// ---------------------------------------------------------------------------
// float4 copy for slicing entity embeddings out of gcn outputs
// ---------------------------------------------------------------------------
__global__ __launch_bounds__(256)
void copy4_kernel(const float* __restrict__ src, float* __restrict__ dst, long n4)
{
    const long i = blockIdx.x * (long)blockDim.x + threadIdx.x;
    if (i < n4) ((float4*)dst)[i] = ((const float4*)src)[i];
}

// ---------------------------------------------------------------------------
extern "C" void kernel_launch(void* const* d_in, const int* in_sizes, int n_in,
                              void* d_out, int out_size, void* d_ws, size_t ws_size,
                              hipStream_t stream)
{
    (void)in_sizes; (void)n_in; (void)out_size; (void)ws_size;
    constexpr int U = 100000, I = 100000, E = 50000, D = 64;
    constexpr int NNZ_UI = 1600000, NNZ_E = 1200000, K = 2;

    const int*   u2i_idx  = (const int*)  d_in[0];
    const float* u2i_val  = (const float*)d_in[1];
    const int*   u2e_idx  = (const int*)  d_in[2];
    const float* u2e_val  = (const float*)d_in[3];
    const int*   i2e_idx  = (const int*)  d_in[4];
    const float* i2e_val  = (const float*)d_in[5];
    const float* u_feat   = (const float*)d_in[6];
    const float* i_feat   = (const float*)d_in[7];
    const float* u2e_feat = (const float*)d_in[8];
    const float* i2e_feat = (const float*)d_in[9];
    const float* Tw_u2i   = (const float*)d_in[10];
    const float* Tb_u2i   = (const float*)d_in[11];
    const float* Iw_u2i   = (const float*)d_in[12];
    const float* Ib_u2i   = (const float*)d_in[13];
    const float* Tw_u2e   = (const float*)d_in[14];
    const float* Tb_u2e   = (const float*)d_in[15];
    const float* Iw_u2e   = (const float*)d_in[16];
    const float* Ib_u2e   = (const float*)d_in[17];
    const float* Tw_i2e   = (const float*)d_in[18];
    const float* Tb_i2e   = (const float*)d_in[19];
    const float* Iw_i2e   = (const float*)d_in[20];
    const float* Ib_i2e   = (const float*)d_in[21];
    const float* uatt_w1  = (const float*)d_in[22];
    const float* uatt_b1  = (const float*)d_in[23];
    const float* uatt_w2  = (const float*)d_in[24];
    const float* iatt_w1  = (const float*)d_in[25];
    const float* iatt_b1  = (const float*)d_in[26];
    const float* iatt_w2  = (const float*)d_in[27];

    float* out = (float*)d_out;
    float* ws  = (float*)d_ws;

    const size_t cellUE = (size_t)(U + E) * D;      // 9.6M floats
    float* tu[2] = { ws,               ws + cellUE     };
    float* ti[2] = { ws + 2 * cellUE,  ws + 3 * cellUE };
    float* tui   = ws + 4 * cellUE;                 // (U+I)*D
    float* pbuf  = tui + (size_t)(U + I) * D;       // scratch, reused per cell

    // ---- u2e cells (x = concat(u_feat, u2e_feat[k]), n = U+E) ----
    for (int k = 0; k < K; ++k) {
        gcn_gemm_kernel<<<(U + E) / 16, 128, 0, stream>>>(
            u_feat, U, u2e_feat + (size_t)k * E * D, E,
            Tw_u2e + k * D * D, Tb_u2e + k * D,
            Iw_u2e + k * D * D, Ib_u2e + k * D, tu[k], pbuf);
        const int* rows = u2e_idx + (size_t)k * 2 * NNZ_E;
        edge_scatter_kernel<<<(4 * NNZ_E + 255) / 256, 256, 0, stream>>>(
            rows, rows + NNZ_E, u2e_val + (size_t)k * NNZ_E, NNZ_E, pbuf, tu[k]);
    }
    // ---- i2e cells (x = concat(i_feat, i2e_feat[k]), n = I+E) ----
    for (int k = 0; k < K; ++k) {
        gcn_gemm_kernel<<<(I + E) / 16, 128, 0, stream>>>(
            i_feat, I, i2e_feat + (size_t)k * E * D, E,
            Tw_i2e + k * D * D, Tb_i2e + k * D,
            Iw_i2e + k * D * D, Ib_i2e + k * D, ti[k], pbuf);
        const int* rows = i2e_idx + (size_t)k * 2 * NNZ_E;
        edge_scatter_kernel<<<(4 * NNZ_E + 255) / 256, 256, 0, stream>>>(
            rows, rows + NNZ_E, i2e_val + (size_t)k * NNZ_E, NNZ_E, pbuf, ti[k]);
    }
    // ---- u2i cell (x = concat(u_feat, i_feat), n = U+I) ----
    gcn_gemm_kernel<<<(U + I) / 16, 128, 0, stream>>>(
        u_feat, U, i_feat, I, Tw_u2i, Tb_u2i, Iw_u2i, Ib_u2i, tui, pbuf);
    edge_scatter_kernel<<<(4 * NNZ_UI + 255) / 256, 256, 0, stream>>>(
        u2i_idx, u2i_idx + NNZ_UI, u2i_val, NNZ_UI, pbuf, tui);

    // ---- relational attention (relations: k=0, k=1, u2i) ----
    rel_att_kernel<<<(U + 7) / 8, 256, 0, stream>>>(
        tu[0], tu[1], tui, U, uatt_w1, uatt_b1, uatt_w2, out);
    rel_att_kernel<<<(I + 7) / 8, 256, 0, stream>>>(
        ti[0], ti[1], tui + (size_t)U * D, I, iatt_w1, iatt_b1, iatt_w2,
        out + (size_t)U * D);

    // ---- entity embedding slices ----
    const size_t embOff = (size_t)(U + I) * D;
    const long   n4     = (long)E * D / 4;
    for (int k = 0; k < K; ++k)
        copy4_kernel<<<(E * D / 4 + 255) / 256, 256, 0, stream>>>(
            tu[k] + (size_t)U * D, out + embOff + (size_t)k * E * D, n4);
    for (int k = 0; k < K; ++k)
        copy4_kernel<<<(E * D / 4 + 255) / 256, 256, 0, stream>>>(
            ti[k] + (size_t)I * D,
            out + embOff + (size_t)K * E * D + (size_t)k * E * D, n4);
}